// CensNet_74302934221370
// MI455X (gfx1250) — compile-verified
//
#include <hip/hip_runtime.h>
#include <hip/hip_bf16.h>
#include <math.h>

// ---------------- problem constants ----------------
#define NN 3000     // nodes
#define MM 6000     // edges
#define FV 128
#define FE 64
#define H1 256
#define H2 128
#define NC 16
#define E2 (2 * MM) // 12000 directed edges in edge_index

// ---------------- WMMA GEMM tiling ----------------
#define BI 128
#define BJ 64
#define KT 32
#define AS_LD (BI + 8)   // 136: (2*AS_LD)%64 == 16 -> lanes16-31 hit different banks
#define BS_LD (BJ + 8)   // 72:  (2*BS_LD)%64 == 16

typedef float v2f __attribute__((ext_vector_type(2)));
typedef float v8f __attribute__((ext_vector_type(8)));

// C[NI,NJ] = opA(A) * opB(B)   with epilogue
//   opA: 0 -> A[i*lda+k] ; 1 -> A[k*lda+i] (transposed source) ; 2 -> A[i*lda+k]*scaleA[k]
//   opB: 0 -> B[k*ldb+j] ; 1 -> B[k*ldb+j]*scaleB[k]
//   epi: 0 -> C=v ; 1 -> C=((i==j)?1:v)*Emask ; 2 -> C=relu(v+bias[j]) ; 3 -> C=v+bias[j]
__launch_bounds__(256)
__global__ void k_wmma_gemm(const float* __restrict__ Asrc, const float* __restrict__ Bsrc,
                            float* __restrict__ C,
                            const float* __restrict__ scaleA, const float* __restrict__ scaleB,
                            const float* __restrict__ Emask, const float* __restrict__ bias,
                            int NI, int NJ, int NK, int lda, int ldb, int ldc,
                            int opA, int opB, int epi)
{
    __shared__ float As[KT * AS_LD];
    __shared__ float Bs[KT * BS_LD];

    const int tid  = threadIdx.x;
    const int lane = tid & 31;
    const int w    = tid >> 5;            // wave 0..7
    const int wi   = (w & 3) * 32;        // wave i offset inside block tile
    const int wj   = (w >> 2) * 32;       // wave j offset
    const int half = lane >> 4;           // 0/1
    const int l16  = lane & 15;

    const int i0 = blockIdx.y * BI;
    const int j0 = blockIdx.x * BJ;

    v8f acc[2][2] = {};

    for (int k0 = 0; k0 < NK; k0 += KT) {
        // ---- stage A tile: As[k][i] ----
        if (opA == 1) {
            #pragma unroll
            for (int t = 0; t < (BI * KT) / 256; ++t) {
                int idx = tid + t * 256;
                int k = idx / BI, i = idx % BI;          // consecutive tid -> consecutive i (coalesced)
                int gi = i0 + i, gk = k0 + k;
                float v = (gi < NI && gk < NK) ? Asrc[(size_t)gk * lda + gi] : 0.0f;
                As[k * AS_LD + i] = v;
            }
        } else {
            #pragma unroll
            for (int t = 0; t < (BI * KT) / 256; ++t) {
                int idx = tid + t * 256;
                int i = idx / KT, k = idx % KT;          // consecutive tid -> consecutive k (coalesced)
                int gi = i0 + i, gk = k0 + k;
                float v = (gi < NI && gk < NK) ? Asrc[(size_t)gi * lda + gk] : 0.0f;
                if (opA == 2 && gk < NK) v *= scaleA[gk];
                As[k * AS_LD + i] = v;
            }
        }
        // ---- stage B tile: Bs[k][j] ----
        #pragma unroll
        for (int t = 0; t < (BJ * KT) / 256; ++t) {
            int idx = tid + t * 256;
            int k = idx / BJ, j = idx % BJ;
            int gj = j0 + j, gk = k0 + k;
            float v = (gj < NJ && gk < NK) ? Bsrc[(size_t)gk * ldb + gj] : 0.0f;
            if (opB == 1 && gk < NK) v *= scaleB[gk];
            Bs[k * BS_LD + j] = v;
        }
        __syncthreads();

        #pragma unroll
        for (int ks = 0; ks < KT; ks += 4) {
            const int kr = ks + 2 * half;   // K = 2*half + r  (r = vgpr index)
            v2f a0, a1, b0, b1;
            a0.x = As[kr * AS_LD + wi + l16];
            a0.y = As[(kr + 1) * AS_LD + wi + l16];
            a1.x = As[kr * AS_LD + wi + 16 + l16];
            a1.y = As[(kr + 1) * AS_LD + wi + 16 + l16];
            b0.x = Bs[kr * BS_LD + wj + l16];
            b0.y = Bs[(kr + 1) * BS_LD + wj + l16];
            b1.x = Bs[kr * BS_LD + wj + 16 + l16];
            b1.y = Bs[(kr + 1) * BS_LD + wj + 16 + l16];
            acc[0][0] = __builtin_amdgcn_wmma_f32_16x16x4_f32(false, a0, false, b0, (short)0, acc[0][0], false, false);
            acc[0][1] = __builtin_amdgcn_wmma_f32_16x16x4_f32(false, a0, false, b1, (short)0, acc[0][1], false, false);
            acc[1][0] = __builtin_amdgcn_wmma_f32_16x16x4_f32(false, a1, false, b0, (short)0, acc[1][0], false, false);
            acc[1][1] = __builtin_amdgcn_wmma_f32_16x16x4_f32(false, a1, false, b1, (short)0, acc[1][1], false, false);
        }
        __syncthreads();
    }

    // ---- epilogue ----
    #pragma unroll
    for (int ti = 0; ti < 2; ++ti)
        #pragma unroll
        for (int tj = 0; tj < 2; ++tj)
            #pragma unroll
            for (int r = 0; r < 8; ++r) {
                int gi = i0 + wi + ti * 16 + r + 8 * half;  // C layout: vgpr r -> M = r / r+8
                int gj = j0 + wj + tj * 16 + l16;
                if (gi < NI && gj < NJ) {
                    float v = acc[ti][tj][r];
                    size_t off = (size_t)gi * ldc + gj;
                    if (epi == 1)      { v = (gi == gj) ? 1.0f : v; C[off] = v * Emask[off]; }
                    else if (epi == 2) { C[off] = fmaxf(v + bias[gj], 0.0f); }
                    else if (epi == 3) { C[off] = v + bias[gj]; }
                    else               { C[off] = v; }
                }
            }
}

// ---------------- small helper kernels ----------------
__global__ void k_fill(float* __restrict__ p, float v, int n) {
    int i = blockIdx.x * blockDim.x + threadIdx.x;
    if (i < n) p[i] = v;
}

__global__ void k_relu(const float* __restrict__ in, float* __restrict__ out, int n) {
    int i = blockIdx.x * blockDim.x + threadIdx.x;
    if (i < n) out[i] = fmaxf(in[i], 0.0f);
}

__global__ void k_deg(const int* __restrict__ dst, float* __restrict__ deg, int E) {
    int e = blockIdx.x * blockDim.x + threadIdx.x;
    if (e < E) atomicAdd(&deg[dst[e]], 1.0f);
}

__global__ void k_dinv(const float* __restrict__ deg, float* __restrict__ dinv, int n) {
    int i = blockIdx.x * blockDim.x + threadIdx.x;
    if (i < n) dinv[i] = rsqrtf(deg[i]);   // deg >= 1 always (self loop)
}

// G[v,f] = XW[v,f]*dinv[v]^2 + b[f]   (self-loop term + bias, scatter base)
__global__ void k_gcn_init(const float* __restrict__ xw, const float* __restrict__ dinv,
                           const float* __restrict__ b, float* __restrict__ G, int n, int F) {
    int idx = blockIdx.x * blockDim.x + threadIdx.x;
    if (idx < n * F) {
        int v = idx / F, f = idx % F;
        float dv = dinv[v];
        G[idx] = xw[idx] * dv * dv + b[f];
    }
}

// G[dst,f] += XW[src,f] * dinv[src]*dinv[dst]
__global__ void k_gcn_scatter(const float* __restrict__ xw, const float* __restrict__ dinv,
                              const int* __restrict__ src, const int* __restrict__ dst,
                              float* __restrict__ G, int E, int F) {
    int t = blockIdx.x * blockDim.x + threadIdx.x;
    if (t < E * F) {
        int e = t / F, f = t % F;
        int s = src[e], d = dst[e];
        atomicAdd(&G[(size_t)d * F + f], xw[(size_t)s * F + f] * dinv[s] * dinv[d]);
    }
}

// out[row] = dot(A[row,:K], p[:K])  — one wave per row
__global__ void k_rowdot(const float* __restrict__ A, const float* __restrict__ p,
                         float* __restrict__ out, int n, int K) {
    int row  = blockIdx.x * 8 + (threadIdx.x >> 5);
    int lane = threadIdx.x & 31;
    if (row >= n) return;                       // uniform per wave
    float s = 0.0f;
    for (int k = lane; k < K; k += 32) s += A[(size_t)row * K + k] * p[k];
    for (int off = 16; off > 0; off >>= 1) s += __shfl_xor(s, off, 32);
    if (lane == 0) out[row] = s;
}

// inv[j] = 1 / max_i A[i,j]  (A row-major rows x cols; threads over j -> coalesced row sweeps)
__global__ void k_colmax_inv(const float* __restrict__ A, float* __restrict__ inv,
                             int rows, int cols) {
    int j = blockIdx.x * blockDim.x + threadIdx.x;
    if (j >= cols) return;
    float m = -3.4e38f;
    for (int i = 0; i < rows; ++i) m = fmaxf(m, A[(size_t)i * cols + j]);
    inv[j] = 1.0f / m;
}

__global__ void k_scale_rows(float* __restrict__ X, const float* __restrict__ s, int total, int F) {
    int idx = blockIdx.x * blockDim.x + threadIdx.x;
    if (idx < total) X[idx] *= s[idx / F];
}

__global__ void k_softmax16(const float* __restrict__ in, float* __restrict__ out, int n) {
    int i = blockIdx.x * blockDim.x + threadIdx.x;
    if (i >= n) return;
    float v[NC];
    float m = -3.4e38f;
    #pragma unroll
    for (int c = 0; c < NC; ++c) { v[c] = in[i * NC + c]; m = fmaxf(m, v[c]); }
    float s = 0.0f;
    #pragma unroll
    for (int c = 0; c < NC; ++c) { v[c] = __expf(v[c] - m); s += v[c]; }
    float r = 1.0f / s;
    #pragma unroll
    for (int c = 0; c < NC; ++c) out[i * NC + c] = v[c] * r;
}

// ---------------- host-side orchestration ----------------
extern "C" void kernel_launch(void* const* d_in, const int* in_sizes, int n_in,
                              void* d_out, int out_size, void* d_ws, size_t ws_size,
                              hipStream_t stream) {
    const float* X     = (const float*)d_in[0];
    const float* Z     = (const float*)d_in[1];
    const float* adj_e = (const float*)d_in[2];
    // d_in[3] adj_v: unused by the reference
    const float* T     = (const float*)d_in[4];
    const int*   ei    = (const int*)d_in[5];
    const int*   src   = ei;            // row 0
    const int*   dst   = ei + E2;       // row 1
    const float* W1  = (const float*)d_in[6];
    const float* b1  = (const float*)d_in[7];
    const float* p2  = (const float*)d_in[8];
    const float* W2  = (const float*)d_in[9];
    const float* b2  = (const float*)d_in[10];
    const float* p3  = (const float*)d_in[11];
    const float* W3  = (const float*)d_in[12];
    const float* b3  = (const float*)d_in[13];
    const float* p32 = (const float*)d_in[14];
    const float* W32 = (const float*)d_in[15];
    const float* b32 = (const float*)d_in[16];
    const float* p4  = (const float*)d_in[17];
    const float* W4  = (const float*)d_in[18];
    const float* b4g = (const float*)d_in[19];
    const float* b4  = (const float*)d_in[20];
    const float* W5  = (const float*)d_in[21];
    const float* b5  = (const float*)d_in[22];
    float* out = (float*)d_out;

    // workspace layout (floats)
    float* ws   = (float*)d_ws;
    float* deg  = ws;                       // 3000
    float* dinv = deg  + NN;                // 3000
    float* xw   = dinv + NN;                // 768000  (max NN*H1)
    float* g    = xw   + NN * H1;           // 768000  Xh / Xh4
    float* g2   = g    + NN * H1;           // 768000  gcn propagate outputs
    float* zh   = g2   + NN * H1;           // 384000  Zh
    float* hew  = zh   + MM * FE;           // 384000  He@W (then row-scaled)
    float* sv   = hew  + MM * FE;           // 3000
    float* se   = sv   + NN;                // 6000
    float* cinv = se   + MM;                // 6000
    float* P    = cinv + MM;                // 768000  T^T @ G  (6000x128)
    float* Abig = P    + MM * H2;           // 36,000,000 (6000x6000)  ~144 MB

    auto gemm = [&](const float* A, const float* B, float* C,
                    int NI, int NJ, int NK, int lda, int ldb, int ldc,
                    int opA, int opB, int epi,
                    const float* sA, const float* sB, const float* E, const float* bias) {
        dim3 grid((NJ + BJ - 1) / BJ, (NI + BI - 1) / BI);
        k_wmma_gemm<<<grid, 256, 0, stream>>>(A, B, C, sA, sB, E, bias,
                                              NI, NJ, NK, lda, ldb, ldc, opA, opB, epi);
    };
    auto blks = [](int n) { return (n + 255) / 256; };

    // ---- GCN normalization (deg includes self loop) ----
    k_fill<<<blks(NN), 256, 0, stream>>>(deg, 1.0f, NN);
    k_deg<<<blks(E2), 256, 0, stream>>>(dst, deg, E2);
    k_dinv<<<blks(NN), 256, 0, stream>>>(deg, dinv, NN);

    // ---- gc1: Xh = relu(GCN(X, W1, b1)) ----
    gemm(X, W1, xw, NN, H1, FV, FV, H1, H1, 0, 0, 0, nullptr, nullptr, nullptr, nullptr);
    k_gcn_init<<<blks(NN * H1), 256, 0, stream>>>(xw, dinv, b1, g, NN, H1);
    k_gcn_scatter<<<blks(E2 * H1), 256, 0, stream>>>(xw, dinv, src, dst, g, E2, H1);
    k_relu<<<blks(NN * H1), 256, 0, stream>>>(g, g, NN * H1);       // Xh
    k_relu<<<blks(MM * FE), 256, 0, stream>>>(Z, zh, MM * FE);      // Zh

    // ---- three edge layers (Xh fixed; only p/W/b differ) ----
    const float* ps[3] = {p2, p3, p32};
    const float* Wk[3] = {W2, W3, W32};
    const float* bk[3] = {b2, b3, b32};
    for (int L = 0; L < 3; ++L) {
        k_rowdot<<<(NN + 7) / 8, 256, 0, stream>>>(g, ps[L], sv, NN, H1);      // sv = Xh @ p^T
        gemm(zh, Wk[L], hew, MM, FE, FE, FE, FE, FE, 0, 0, 0,
             nullptr, nullptr, nullptr, nullptr);                               // HeW = Zh @ W
        // A = (T^T diag(sv) T  with diag->1) * adj_e      [the 216 GFLOP GEMM]
        gemm(T, T, Abig, MM, MM, NN, MM, MM, MM, 1, 1, 1,
             nullptr, sv, adj_e, nullptr);
        k_colmax_inv<<<blks(MM), 256, 0, stream>>>(Abig, cinv, MM, MM);
        k_scale_rows<<<blks(MM * FE), 256, 0, stream>>>(hew, cinv, MM * FE, FE); // fold colmax into HeW
        gemm(Abig, hew, zh, MM, FE, MM, MM, FE, FE, 0, 0, 2,
             nullptr, nullptr, nullptr, bk[L]);                                  // Zh = relu(A@HeWs + b)
    }

    // ---- gc4: X4 = T diag(se) (T^T @ GCN(Xh,W4,b4g)) + b4 ; Xh4 = relu(X4) ----
    k_rowdot<<<(MM + 7) / 8, 256, 0, stream>>>(zh, p4, se, MM, FE);             // se = Zh @ p4^T
    gemm(g, W4, xw, NN, H2, H1, H1, H2, H2, 0, 0, 0, nullptr, nullptr, nullptr, nullptr);
    k_gcn_init<<<blks(NN * H2), 256, 0, stream>>>(xw, dinv, b4g, g2, NN, H2);
    k_gcn_scatter<<<blks(E2 * H2), 256, 0, stream>>>(xw, dinv, src, dst, g2, E2, H2);
    gemm(T, g2, P, MM, H2, NN, MM, H2, H2, 1, 0, 0,
         nullptr, nullptr, nullptr, nullptr);                                    // P = T^T @ G
    gemm(T, P, g, NN, H2, MM, MM, H2, H2, 2, 0, 2,
         se, nullptr, nullptr, b4);                                              // Xh4 = relu((T*se)@P + b4)

    // ---- gc5: softmax(GCN(Xh4, W5, b5)) ----
    gemm(g, W5, xw, NN, NC, H2, H2, NC, NC, 0, 0, 0, nullptr, nullptr, nullptr, nullptr);
    k_gcn_init<<<blks(NN * NC), 256, 0, stream>>>(xw, dinv, b5, g2, NN, NC);
    k_gcn_scatter<<<blks(E2 * NC), 256, 0, stream>>>(xw, dinv, src, dst, g2, E2, NC);
    k_softmax16<<<blks(NN), 256, 0, stream>>>(g2, out, NN);
}